// BruteForce_83099027243251
// MI455X (gfx1250) — compile-verified
//
#include <hip/hip_runtime.h>

// ---------------- problem constants (match reference) ----------------
constexpr int   DIMS   = 128;      // D
constexpr int   NQ     = 512;      // B
constexpr int   NCAND  = 500000;   // N
constexpr int   K_TOP  = 100;      // K
// ---------------- algorithm tuning ----------------
constexpr int   NBINS   = 2048;    // 11-bit radix of monotonic float key
constexpr int   QTILE   = 16;      // queries per workgroup (one WMMA M tile)
constexpr int   WAVES   = 8;       // 256 threads, wave32
constexpr int   NTILE   = WAVES * 16;              // 128 candidates / iter
constexpr int   NCHUNKS = 16;                      // grid.y
constexpr int   CHUNK   = (NCAND + NCHUNKS - 1) / NCHUNKS;  // 31250
constexpr int   ITERS   = (CHUNK + NTILE - 1) / NTILE;      // 245
constexpr int   CAP     = 1024;    // collected candidates per query (>= K)
constexpr int   SORTN   = 1024;    // bitonic sort size (pow2 >= CAP)

typedef float v2f __attribute__((ext_vector_type(2)));
typedef float v8f __attribute__((ext_vector_type(8)));

struct ScorePair { float s; int id; };

// monotonic mapping: float total order -> unsigned total order
__device__ __forceinline__ unsigned mkey(float f) {
    unsigned u = __float_as_uint(f);
    return (u & 0x80000000u) ? ~u : (u | 0x80000000u);
}
__device__ __forceinline__ float unmkey(unsigned k) {
    unsigned u = (k & 0x80000000u) ? (k & 0x7FFFFFFFu) : ~k;
    return __uint_as_float(u);
}

// ---------------------------------------------------------------
// zero scratch
__global__ void zero_u32(unsigned* __restrict__ p, int n) {
    int i = blockIdx.x * blockDim.x + threadIdx.x;
    if (i < n) p[i] = 0u;
}

// ---------------------------------------------------------------
// Fused GEMM (fp32 WMMA 16x16x4) + histogram (MODE 0) / collect (MODE 1).
// Workgroup: 16-query tile (blockIdx.x) x candidate chunk (blockIdx.y).
// Per wave: one 16-candidate N tile per iteration; A held in VGPRs.
template <int MODE>
__global__ __launch_bounds__(256)
void score_pass(const float* __restrict__ Q, const float* __restrict__ C,
                unsigned*   __restrict__ ghist,   // [NQ][NBINS]   (MODE 0)
                const int2* __restrict__ qinfo,   // [NQ] {thrBin, above} (MODE 1)
                unsigned*   __restrict__ cnt,     // [NQ]          (MODE 1)
                ScorePair*  __restrict__ buf)     // [NQ][CAP]     (MODE 1)
{
    __shared__ unsigned lhist[(MODE == 0) ? (QTILE * NBINS) : 1];

    const int tid  = threadIdx.x;
    const int wave = tid >> 5;
    const int lane = tid & 31;
    const int qrow = lane & 15;     // row index inside 16-tile (A) == N col (B/C/D)
    const int khalf = lane >> 4;    // 0: K={0,1} of 4-block, 1: K={2,3}
    const int qbase = blockIdx.x * QTILE;
    const int chunkStart = blockIdx.y * CHUNK;
    const int nEnd = (chunkStart + CHUNK < NCAND) ? (chunkStart + CHUNK) : NCAND;

    if (MODE == 0) {
        for (int i = tid; i < QTILE * NBINS; i += blockDim.x) lhist[i] = 0u;
        __syncthreads();
    }

    // ---- preload A (query rows) into VGPRs: 32 K-steps x 2 floats ----
    v2f av[32];
    {
        const float* qptr = Q + (size_t)(qbase + qrow) * DIMS;
        #pragma unroll
        for (int j = 0; j < 32; ++j) {
            const float2 q2 = *reinterpret_cast<const float2*>(qptr + 4 * j + 2 * khalf);
            av[j].x = q2.x; av[j].y = q2.y;
        }
    }

    int thr[8];
    if (MODE == 1) {
        #pragma unroll
        for (int r = 0; r < 8; ++r)
            thr[r] = qinfo[qbase + r + khalf * 8].x;
    }

    for (int it = 0; it < ITERS; ++it) {
        const int n_base = chunkStart + it * NTILE + wave * 16;
        const int n      = n_base + qrow;                    // this lane's B row / C col
        const int nclamp = (n < NCAND) ? n : (NCAND - 1);    // OOB cols discarded later
        const float* crow = C + (size_t)nclamp * DIMS;

        v8f acc = {0.f, 0.f, 0.f, 0.f, 0.f, 0.f, 0.f, 0.f};
        #pragma unroll
        for (int j = 0; j < 32; ++j) {
            const float2 c2 = *reinterpret_cast<const float2*>(crow + 4 * j + 2 * khalf);
            v2f bv; bv.x = c2.x; bv.y = c2.y;
            acc = __builtin_amdgcn_wmma_f32_16x16x4_f32(
                      false, av[j], false, bv, (short)0, acc, false, false);
        }

        if (n < nEnd) {   // per-lane predicate only around scalar post-processing
            #pragma unroll
            for (int r = 0; r < 8; ++r) {
                const float s = acc[r];
                const unsigned key = mkey(s);
                const int bin = (int)(key >> 21);
                const int mq  = r + khalf * 8;   // query row within tile
                if (MODE == 0) {
                    atomicAdd(&lhist[mq * NBINS + bin], 1u);
                } else {
                    if (bin >= thr[r]) {
                        const int qg = qbase + mq;
                        unsigned pos = atomicAdd(&cnt[qg], 1u);
                        if (pos < (unsigned)CAP) {
                            buf[(size_t)qg * CAP + pos] = ScorePair{s, n};
                        }
                    }
                }
            }
        }
    }

    if (MODE == 0) {
        __syncthreads();
        for (int i = tid; i < QTILE * NBINS; i += blockDim.x) {
            const unsigned v = lhist[i];
            if (v)
                atomicAdd(&ghist[(size_t)(qbase + (i >> 11)) * NBINS + (i & (NBINS - 1))], v);
        }
    }
}

// ---------------------------------------------------------------
// Per-query threshold bin: largest bin T s.t. count(bins >= T) >= K.
__global__ void find_threshold(const unsigned* __restrict__ ghist,
                               int2* __restrict__ qinfo)
{
    const int q = blockIdx.x;
    if (threadIdx.x != 0) return;
    const unsigned* h = ghist + (size_t)q * NBINS;
    unsigned cum = 0; int T = 0; int above = 0;
    for (int b = NBINS - 1; b >= 0; --b) {
        const unsigned c = h[b];
        if (cum + c >= (unsigned)K_TOP) { T = b; above = (int)cum; break; }
        cum += c;
    }
    qinfo[q] = make_int2(T, above);
}

// ---------------------------------------------------------------
// Per-query bitonic sort of collected candidates; emit top-K (desc, stable).
__global__ __launch_bounds__(256)
void topk_sort(const unsigned* __restrict__ cnt,
               const ScorePair* __restrict__ buf,
               const int* __restrict__ identifiers,
               float* __restrict__ out_vals,
               float* __restrict__ out_ids)
{
    __shared__ unsigned skey[SORTN];
    __shared__ int      sid[SORTN];

    const int q = blockIdx.x;
    const int nIn = min((int)cnt[q], CAP);
    const ScorePair* b = buf + (size_t)q * CAP;

    for (int i = threadIdx.x; i < SORTN; i += blockDim.x) {
        if (i < nIn) { skey[i] = mkey(b[i].s); sid[i] = b[i].id; }
        else         { skey[i] = 0u;           sid[i] = 0x7FFFFFFF; }
    }
    __syncthreads();

    // bitonic sort: final order = descending key, ascending id on ties
    for (int k = 2; k <= SORTN; k <<= 1) {
        for (int j = k >> 1; j > 0; j >>= 1) {
            for (int i = threadIdx.x; i < SORTN; i += blockDim.x) {
                const int ixj = i ^ j;
                if (ixj > i) {
                    const unsigned ka = skey[i], kb = skey[ixj];
                    const int ia = sid[i], ib = sid[ixj];
                    const bool up = ((i & k) == 0);
                    const bool precBA = (kb > ka) || (kb == ka && ib < ia);
                    const bool precAB = (ka > kb) || (ka == kb && ia < ib);
                    if (up ? precBA : precAB) {
                        skey[i] = kb; skey[ixj] = ka;
                        sid[i]  = ib; sid[ixj]  = ia;
                    }
                }
            }
            __syncthreads();
        }
    }

    for (int i = threadIdx.x; i < K_TOP; i += blockDim.x) {
        out_vals[(size_t)q * K_TOP + i] = unmkey(skey[i]);
        const int idx = sid[i];
        const int ident = (idx >= 0 && idx < NCAND) ? identifiers[idx] : idx;
        out_ids[(size_t)q * K_TOP + i] = (float)ident;
    }
}

// ---------------------------------------------------------------
extern "C" void kernel_launch(void* const* d_in, const int* in_sizes, int n_in,
                              void* d_out, int out_size, void* d_ws, size_t ws_size,
                              hipStream_t stream)
{
    const float* Q   = (const float*)d_in[0];   // [512,128]
    const float* C   = (const float*)d_in[1];   // [500000,128]
    const int*   ids = (const int*)d_in[2];     // [500000]
    (void)in_sizes; (void)n_in; (void)out_size; (void)ws_size;

    float* out_vals = (float*)d_out;                         // [512*100]
    float* out_ids  = out_vals + (size_t)NQ * K_TOP;         // [512*100]

    // workspace layout (~8.4 MB total)
    unsigned*  hist  = (unsigned*)d_ws;                      // 512*2048 u32 = 4 MB
    unsigned*  cnt   = hist + (size_t)NQ * NBINS;            // 512 u32
    int2*      qinfo = (int2*)(cnt + NQ);                    // 512 int2
    ScorePair* buf   = (ScorePair*)(qinfo + NQ);             // 512*1024*8 B = 4 MB

    const int nz = NQ * NBINS + NQ;                          // hist + cnt (contiguous)
    zero_u32<<<(nz + 255) / 256, 256, 0, stream>>>(hist, nz);

    dim3 grid(NQ / QTILE, NCHUNKS);                          // 32 x 16 workgroups
    score_pass<0><<<grid, 256, 0, stream>>>(Q, C, hist, nullptr, nullptr, nullptr);
    find_threshold<<<NQ, 32, 0, stream>>>(hist, qinfo);
    score_pass<1><<<grid, 256, 0, stream>>>(Q, C, nullptr, qinfo, cnt, buf);
    topk_sort<<<NQ, 256, 0, stream>>>(cnt, buf, ids, out_vals, out_ids);
}